// FullyInteractedLinearNextPredictor_1700807049321
// MI455X (gfx1250) — compile-verified
//
#include <hip/hip_runtime.h>

// CDNA5 / gfx1250 fused kernel: out = x * (x @ W.T) + bias
// GEMM via V_WMMA_F32_16X16X4_F32 (native f32 WMMA, wave32),
// LDS staging via Tensor Data Mover (tensor_load_to_lds) with double buffering.

typedef float v2f __attribute__((ext_vector_type(2)));
typedef float v8f __attribute__((ext_vector_type(8)));
typedef unsigned int u32x4 __attribute__((ext_vector_type(4)));
typedef int i32x4 __attribute__((ext_vector_type(4)));
typedef int i32x8 __attribute__((ext_vector_type(8)));

#define LATENT 512
#define BATCH  1024
#define BM     64    // rows per workgroup block (4 waves x 16)
#define BN     64    // cols per workgroup block
#define KC     32    // K chunk staged through LDS (per buffer)
#define LSTR   36    // LDS row stride in floats (32 + 4 pad: 16B-aligned, conflict-free b64)
#define NCHUNK (LATENT / KC)

#if defined(__gfx1250__) && __has_builtin(__builtin_amdgcn_tensor_load_to_lds) && (__clang_major__ <= 22)
#define USE_TDM 1
#else
#define USE_TDM 0
#endif

#if USE_TDM
// Build a Tensor DMA Descriptor (D#) for a 2D tile:
//   tile_dim0 = KC elements (f32) along the contiguous dim, tile_dim1 = tile_rows rows,
//   tensor row stride = LATENT, LDS rows padded by 4 DWORDs every 32 DWORDs (-> LSTR).
__device__ __forceinline__ void tdm_load_tile(unsigned lds_byte_addr,
                                              const float* gptr,
                                              unsigned tensor_rows,
                                              unsigned tile_rows)
{
    unsigned long long ga = (unsigned long long)(size_t)gptr;
    u32x4 g0;
    g0[0] = 1u;                                               // count=1, user D#, no gather
    g0[1] = lds_byte_addr;                                    // lds_addr [63:32]
    g0[2] = (unsigned)(ga & 0xFFFFFFFFull);                   // global_addr [95:64]
    g0[3] = (unsigned)((ga >> 32) & 0x1FFFFFFull) | (2u << 30); // global_addr[56:32] + type=2
    i32x8 g1;
    g1[0] = (2 << 16)        // data_size = 4 bytes
          | (1 << 20)        // pad_enable
          | (4 << 22)        // pad_interval: 32 DWORDs (= KC floats, one row)
          | (3 << 25);       // pad_amount: 4 DWORDs (-> row stride LSTR = 36 floats)
    g1[1] = (int)((LATENT & 0xFFFF) << 16);                   // tensor_dim0[15:0] -> bits[63:48]
    g1[2] = (int)(((LATENT >> 16) & 0xFFFF) | ((tensor_rows & 0xFFFF) << 16)); // dim0 hi | dim1 lo
    g1[3] = (int)(((tensor_rows >> 16) & 0xFFFF) | (KC << 16)); // dim1 hi | tile_dim0
    g1[4] = (int)tile_rows;                                   // tile_dim1 (tile_dim2 = 0)
    g1[5] = LATENT;                                           // tensor_dim0_stride[31:0]
    g1[6] = 0;                                                // stride0 hi | stride1 lo (2D)
    g1[7] = 0;
    i32x4 gz = {0, 0, 0, 0};                                  // groups 2/3 unused (2D tensor)
    __builtin_amdgcn_tensor_load_to_lds(g0, g1, gz, gz, 0);
}
#endif

__global__ __launch_bounds__(128)
void fused_xwx_bias_kernel(const float* __restrict__ x,
                           const float* __restrict__ w,
                           const float* __restrict__ bias,
                           float* __restrict__ out)
{
    __shared__ float xs[2][BM * LSTR];   // 2 x 9216 B
    __shared__ float wsh[2][BN * LSTR];  // 2 x 9216 B   (36 KB total)

    const int tid    = threadIdx.x;
    const int lane   = tid & 31;
    const int wave   = tid >> 5;        // 0..3 -> 16-row strip of the 64-row block
    const int lane16 = lane & 15;
    const int half   = lane >> 4;       // K-half select per ISA A/B fragment layout
    const int mBlock = blockIdx.x;      // 0..15
    const int nBlock = blockIdx.y;      // 0..7

    const float* xTileBase = x + (size_t)(mBlock * BM) * LATENT;
    const float* wTileBase = w + (size_t)(nBlock * BN) * LATENT;

    v8f acc[4] = {};                    // 4 N-tiles of 16x16 f32 accumulators

#if USE_TDM
    if (wave == 0) {                    // preload chunk 0 into buffer 0 (wave-uniform branch)
        tdm_load_tile((unsigned)(size_t)(void*)&xs[0][0],  xTileBase, BATCH,  BM);
        tdm_load_tile((unsigned)(size_t)(void*)&wsh[0][0], wTileBase, LATENT, BN);
    }
#endif

    for (int c = 0; c < NCHUNK; ++c) {
        const int k0  = c * KC;
        const int cur = c & 1;

#if USE_TDM
        if (wave == 0) {
            if (c + 1 < NCHUNK) {
                // issue next chunk's DMA into the other buffer, then wait only for
                // the current pair (in-order completion -> TENSORcnt <= 2)
                tdm_load_tile((unsigned)(size_t)(void*)&xs[cur ^ 1][0],
                              xTileBase + k0 + KC, BATCH, BM);
                tdm_load_tile((unsigned)(size_t)(void*)&wsh[cur ^ 1][0],
                              wTileBase + k0 + KC, LATENT, BN);
                __builtin_amdgcn_s_wait_tensorcnt((short)2);
            } else {
                __builtin_amdgcn_s_wait_tensorcnt((short)0);
            }
        }
#else
        // ---- manual staging fallback: 64 x 32 floats each = 512 float4, 4/thread ----
        {
            const float4* gx = (const float4*)(xTileBase + k0);
            #pragma unroll
            for (int t = 0; t < 4; ++t) {
                int f  = tid + t * 128;          // 0..511
                int r  = f >> 3;                 // row 0..63
                int c4 = f & 7;                  // float4 col 0..7
                float4 v = gx[(size_t)r * (LATENT / 4) + c4];
                float* dst = &xs[cur][r * LSTR + c4 * 4];
                dst[0] = v.x; dst[1] = v.y; dst[2] = v.z; dst[3] = v.w;
            }
            const float4* gw = (const float4*)(wTileBase + k0);
            #pragma unroll
            for (int t = 0; t < 4; ++t) {
                int f  = tid + t * 128;
                int r  = f >> 3;
                int c4 = f & 7;
                float4 v = gw[(size_t)r * (LATENT / 4) + c4];
                float* dst = &wsh[cur][r * LSTR + c4 * 4];
                dst[0] = v.x; dst[1] = v.y; dst[2] = v.z; dst[3] = v.w;
            }
            if (k0 + KC < LATENT) {              // global_prefetch_b8 of next chunk
                __builtin_prefetch(xTileBase + (size_t)(tid >> 1) * LATENT + k0 + KC, 0, 3);
                __builtin_prefetch(wTileBase + (size_t)(tid >> 1) * LATENT + k0 + KC, 0, 3);
            }
        }
#endif
        __syncthreads();

        // ---- compute: 8 k-steps of V_WMMA_F32_16X16X4_F32, 4 N-tiles each ----
        // A tile 16x4: lane L holds row M=L%16, cols K=(L/16)*2+{0,1}
        // B tile 4x16: lane L holds col N=L%16, rows K=(L/16)*2+{0,1}  (B[k][n] = W[n][k])
        const float* xrow = &xs[cur][(wave * 16 + lane16) * LSTR];
        const float* wbase = &wsh[cur][lane16 * LSTR];
        #pragma unroll
        for (int kk = 0; kk < KC; kk += 4) {
            const int col = kk + half * 2;
            v2f a = *(const v2f*)(xrow + col);
            #pragma unroll
            for (int t = 0; t < 4; ++t) {
                v2f b = *(const v2f*)(wbase + t * 16 * LSTR + col);
                acc[t] = __builtin_amdgcn_wmma_f32_16x16x4_f32(
                    /*neg_a=*/false, a, /*neg_b=*/false, b,
                    /*c_mod=*/(short)0, acc[t],
                    /*reuse_a=*/false, /*reuse_b=*/false);
            }
        }
        __syncthreads();
    }

    // ---- fused epilogue: out[b,i] = x[b,i] * wx[b,i] + bias[i] ----
    // C/D layout: VGPR v, lane L -> M = v + (L/16)*8, N = L%16
    const int gmBase = mBlock * BM + wave * 16 + half * 8;
    const int gnBase = nBlock * BN;
    #pragma unroll
    for (int t = 0; t < 4; ++t) {
        const int gi = gnBase + t * 16 + lane16;
        const float bi = bias[gi];
        #pragma unroll
        for (int v = 0; v < 8; ++v) {
            const int gm = gmBase + v;
            const size_t idx = (size_t)gm * LATENT + gi;
            out[idx] = x[idx] * acc[t][v] + bi;
        }
    }
}

extern "C" void kernel_launch(void* const* d_in, const int* in_sizes, int n_in,
                              void* d_out, int out_size, void* d_ws, size_t ws_size,
                              hipStream_t stream) {
    const float* x    = (const float*)d_in[0];  // (1024, 512)
    const float* w    = (const float*)d_in[1];  // (512, 512)
    const float* bias = (const float*)d_in[2];  // (512,)
    float* out = (float*)d_out;                 // (1024, 512)

    dim3 grid(BATCH / BM, LATENT / BN);         // 16 x 8 = 128 workgroups
    dim3 block(128);                            // 4 waves (wave32)
    fused_xwx_bias_kernel<<<grid, block, 0, stream>>>(x, w, bias, out);
}